// LinkPredictionGCN_48859547959538
// MI455X (gfx1250) — compile-verified
//
#include <hip/hip_runtime.h>
#include <hip/hip_bf16.h>

typedef __attribute__((ext_vector_type(2))) float v2f;
typedef __attribute__((ext_vector_type(8))) float v8f;

// ---------------------------------------------------------------------------
// Degree / normalization
// ---------------------------------------------------------------------------
__global__ void zero_f32_kernel(float* __restrict__ p, int n) {
    int i = blockIdx.x * blockDim.x + threadIdx.x;
    if (i < n) p[i] = 0.0f;
}

__global__ void count_deg_kernel(const int* __restrict__ dst, float* __restrict__ deg, int n_edges) {
    int e = blockIdx.x * blockDim.x + threadIdx.x;
    if (e < n_edges) atomicAdd(&deg[dst[e]], 1.0f);
}

__global__ void dinv_kernel(float* __restrict__ deg, float* __restrict__ dinv, int n_nodes) {
    int i = blockIdx.x * blockDim.x + threadIdx.x;
    if (i < n_nodes) dinv[i] = rsqrtf(deg[i] + 1.0f);  // +1 for self loop
}

// ---------------------------------------------------------------------------
// Layer-1 GEMM: H = X @ W1   (X: [N,128] row-major, W1: [128,128] row-major)
// fp32 WMMA 16x16x4. Block = 256 threads = 8 waves; W1 staged in LDS (64KB);
// each wave owns a 16-col slice and walks 8 row-tiles (128 rows / block).
// ---------------------------------------------------------------------------
__global__ __launch_bounds__(256) void gemm1_wmma_kernel(
    const float* __restrict__ X, const float* __restrict__ W,
    float* __restrict__ H, int n_rows) {
    __shared__ float lw[128 * 128];  // 64 KB

    const int tid = threadIdx.x;
    // Cooperative load of W1 into LDS (float4, fully coalesced)
    const float4* Wv = (const float4*)W;
    float4* lwv = (float4*)lw;
    #pragma unroll
    for (int i = 0; i < 16; ++i) lwv[tid + i * 256] = Wv[tid + i * 256];
    __syncthreads();

    const int wave = tid >> 5;      // 0..7 -> output column slice
    const int lane = tid & 31;
    const int half = lane >> 4;     // 0: lanes 0-15, 1: lanes 16-31
    const int l16  = lane & 15;
    const int col0 = wave * 16;

    const int tile_base = blockIdx.x * 8;
    for (int t = 0; t < 8; ++t) {
        const int row0 = (tile_base + t) * 16;
        if (row0 + 16 > n_rows) break;  // wave-uniform: EXEC stays all-ones

        // Prefetch next row-tile of X while we compute this one
        if (row0 + 32 <= n_rows)
            __builtin_prefetch(X + (row0 + 16 + l16) * 128, 0, 0);

        v8f c = {};
        // A fragment source: row M=l16 of this tile; lane half selects K pair
        const float* xrow = X + (size_t)(row0 + l16) * 128;
        #pragma unroll
        for (int k = 0; k < 128; k += 4) {
            const int kk = k + half * 2;
            // A 16x4 (f32): lanes 0-15 -> K={k,k+1}, lanes 16-31 -> K={k+2,k+3}
            v2f a = *(const v2f*)(xrow + kk);  // 8B-aligned float2 load
            // B 4x16 (f32): V0/V1 hold rows {kk, kk+1} at N = col0+l16
            v2f b;
            b.x = lw[kk * 128 + col0 + l16];
            b.y = lw[(kk + 1) * 128 + col0 + l16];
            c = __builtin_amdgcn_wmma_f32_16x16x4_f32(
                    false, a, false, b, (short)0, c, false, false);
        }
        // C/D layout: VGPR r -> M = r (lanes 0-15) / r+8 (lanes 16-31), N = l16
        float* hout = H + (size_t)(row0 + half * 8) * 128 + col0 + l16;
        #pragma unroll
        for (int r = 0; r < 8; ++r) hout[r * 128] = c[r];
    }
}

// ---------------------------------------------------------------------------
// out1[i][:] = h[i][:] * dinv[i]^2 + b1   (self-loop term + bias, pre-relu)
// ---------------------------------------------------------------------------
__global__ void init_out1_kernel(const float* __restrict__ H,
                                 const float* __restrict__ dinv,
                                 const float* __restrict__ b1,
                                 float* __restrict__ out, int n_nodes) {
    int idx = blockIdx.x * blockDim.x + threadIdx.x;  // one float4 per thread
    int n = idx >> 5;
    int q = idx & 31;
    if (n >= n_nodes) return;
    float di = dinv[n];
    float self = di * di;
    float4 h = ((const float4*)(H + (size_t)n * 128))[q];
    float4 b = ((const float4*)b1)[q];
    float4 o;
    o.x = h.x * self + b.x;
    o.y = h.y * self + b.y;
    o.z = h.z * self + b.z;
    o.w = h.w * self + b.w;
    ((float4*)(out + (size_t)n * 128))[q] = o;
}

// ---------------------------------------------------------------------------
// Edge aggregation layer 1: out1[dst] += h[src] * dinv[src]*dinv[dst]
// One wave per edge; each lane handles 4 channels (float4 gather + 4 atomics)
// ---------------------------------------------------------------------------
__global__ __launch_bounds__(256) void edge_agg1_kernel(
    const float* __restrict__ H, const int* __restrict__ src,
    const int* __restrict__ dst, const float* __restrict__ dinv,
    float* __restrict__ out, int n_edges) {
    const int wave = threadIdx.x >> 5;
    const int lane = threadIdx.x & 31;
    const int e = blockIdx.x * 8 + wave;
    if (e >= n_edges) return;
    const int s = src[e];
    const int d = dst[e];
    const float norm = dinv[s] * dinv[d];
    float4 v = ((const float4*)(H + (size_t)s * 128))[lane];
    float* o = out + (size_t)d * 128 + lane * 4;
    atomicAdd(o + 0, v.x * norm);
    atomicAdd(o + 1, v.y * norm);
    atomicAdd(o + 2, v.z * norm);
    atomicAdd(o + 3, v.w * norm);
}

// ---------------------------------------------------------------------------
// Layer-2 matvec: zlin[n] = relu(out1[n][:]) . W2   (128 -> 1)
// One wave per node; 4 channels/lane; wave32 butterfly reduce.
// ---------------------------------------------------------------------------
__global__ __launch_bounds__(256) void matvec2_kernel(
    const float* __restrict__ O1, const float* __restrict__ W2,
    float* __restrict__ zlin, int n_nodes) {
    const int wave = threadIdx.x >> 5;
    const int lane = threadIdx.x & 31;
    const int n = blockIdx.x * 8 + wave;
    if (n >= n_nodes) return;
    float4 v = ((const float4*)(O1 + (size_t)n * 128))[lane];
    float4 w = ((const float4*)W2)[lane];
    float s = fmaxf(v.x, 0.0f) * w.x + fmaxf(v.y, 0.0f) * w.y +
              fmaxf(v.z, 0.0f) * w.z + fmaxf(v.w, 0.0f) * w.w;
    #pragma unroll
    for (int off = 16; off > 0; off >>= 1) s += __shfl_xor(s, off, 32);
    if (lane == 0) zlin[n] = s;
}

// ---------------------------------------------------------------------------
// z2[i] = zlin[i] * dinv[i]^2 + b2
// ---------------------------------------------------------------------------
__global__ void init_z2_kernel(const float* __restrict__ zlin,
                               const float* __restrict__ dinv,
                               const float* __restrict__ b2,
                               float* __restrict__ z2, int n_nodes) {
    int i = blockIdx.x * blockDim.x + threadIdx.x;
    if (i >= n_nodes) return;
    float di = dinv[i];
    z2[i] = zlin[i] * (di * di) + b2[0];
}

// ---------------------------------------------------------------------------
// Edge aggregation layer 2 (scalar): z2[dst] += zlin[src] * norm
// ---------------------------------------------------------------------------
__global__ void edge_agg2_kernel(const float* __restrict__ zlin,
                                 const int* __restrict__ src,
                                 const int* __restrict__ dst,
                                 const float* __restrict__ dinv,
                                 float* __restrict__ z2, int n_edges) {
    int e = blockIdx.x * blockDim.x + threadIdx.x;
    if (e >= n_edges) return;
    int s = src[e], d = dst[e];
    atomicAdd(&z2[d], zlin[s] * dinv[s] * dinv[d]);
}

// ---------------------------------------------------------------------------
// Decode: out[e] = sigmoid(z2[src] * z2[dst])
// ---------------------------------------------------------------------------
__global__ void decode_kernel(const float* __restrict__ z2,
                              const int* __restrict__ src,
                              const int* __restrict__ dst,
                              float* __restrict__ out, int n_edges) {
    int e = blockIdx.x * blockDim.x + threadIdx.x;
    if (e >= n_edges) return;
    float l = z2[src[e]] * z2[dst[e]];
    out[e] = 1.0f / (1.0f + expf(-l));
}

// ---------------------------------------------------------------------------
extern "C" void kernel_launch(void* const* d_in, const int* in_sizes, int n_in,
                              void* d_out, int out_size, void* d_ws, size_t ws_size,
                              hipStream_t stream) {
    const float* x  = (const float*)d_in[0];
    const int*   ei = (const int*)d_in[1];
    const float* W1 = (const float*)d_in[2];
    const float* b1 = (const float*)d_in[3];
    const float* W2 = (const float*)d_in[4];
    const float* b2 = (const float*)d_in[5];

    const int CH = 128;
    const int N  = in_sizes[0] / CH;   // 50000
    const int E  = in_sizes[1] / 2;    // 800000
    const int* src = ei;
    const int* dst = ei + E;

    // Workspace layout (all offsets 256B-aligned)
    char* ws = (char*)d_ws;
    size_t off = 0;
    float* h    = (float*)(ws + off); off += (size_t)N * CH * sizeof(float);   // 25.6 MB
    float* o1   = (float*)(ws + off); off += (size_t)N * CH * sizeof(float);   // 25.6 MB
    float* deg  = (float*)(ws + off); off += ((size_t)N * sizeof(float) + 255) & ~(size_t)255;
    float* dinv = (float*)(ws + off); off += ((size_t)N * sizeof(float) + 255) & ~(size_t)255;
    float* zlin = (float*)(ws + off); off += ((size_t)N * sizeof(float) + 255) & ~(size_t)255;
    float* z2   = (float*)(ws + off); off += ((size_t)N * sizeof(float) + 255) & ~(size_t)255;
    float* out  = (float*)d_out;

    const int B = 256;
    // 1) degrees and dinv
    zero_f32_kernel<<<(N + B - 1) / B, B, 0, stream>>>(deg, N);
    count_deg_kernel<<<(E + B - 1) / B, B, 0, stream>>>(dst, deg, E);
    dinv_kernel<<<(N + B - 1) / B, B, 0, stream>>>(deg, dinv, N);

    // 2) H = X @ W1 (fp32 WMMA); 8 row-tiles (128 rows) per block
    int ntiles = N / 16;                         // 3125 (N divisible by 16)
    gemm1_wmma_kernel<<<(ntiles + 7) / 8, B, 0, stream>>>(x, W1, h, N);

    // 3) self-loop + bias init, then edge scatter-add
    init_out1_kernel<<<((N * 32) + B - 1) / B, B, 0, stream>>>(h, dinv, b1, o1, N);
    edge_agg1_kernel<<<(E + 7) / 8, B, 0, stream>>>(h, src, dst, dinv, o1, E);

    // 4) layer-2: relu + matvec (128 -> 1)
    matvec2_kernel<<<(N + 7) / 8, B, 0, stream>>>(o1, W2, zlin, N);

    // 5) scalar aggregation
    init_z2_kernel<<<(N + B - 1) / B, B, 0, stream>>>(zlin, dinv, b2, z2, N);
    edge_agg2_kernel<<<(E + B - 1) / B, B, 0, stream>>>(zlin, src, dst, dinv, z2, E);

    // 6) edgewise decode
    decode_kernel<<<(E + B - 1) / B, B, 0, stream>>>(z2, src, dst, out, E);
}